// Cluster_10694468567403
// MI455X (gfx1250) — compile-verified
//
#include <hip/hip_runtime.h>
#include <stdint.h>

typedef float v2f __attribute__((ext_vector_type(2)));
typedef float v8f __attribute__((ext_vector_type(8)));
typedef unsigned int u32x4 __attribute__((ext_vector_type(4)));
typedef int i32x4 __attribute__((ext_vector_type(4)));
typedef int i32x8 __attribute__((ext_vector_type(8)));

#define DIM 128
#define NCENT 256
#define WAVES_PER_WG 8
#define ROWS_PER_WG (WAVES_PER_WG * 16)
#define CTILE 16                     // centers per column tile
#define NTILES (NCENT / CTILE)       // 16
#define TILE_FLOATS (CTILE * DIM)    // 2048 floats = 8KB

struct ClusterWs {
  unsigned long long argmin[NCENT]; // packed (dist_bits<<32)|row  (dist>=0 => monotonic)
  float c2[NCENT];                  // ||center_k||^2
  float loss;                       // accumulated sum of per-row min distances
};

// ---------------------------------------------------------------------------
// TDM: DMA one 16x128 f32 tile of `centers` into LDS (Tensor Data Mover, D#
// built per CDNA5 ISA ch.8; tracked by TENSORcnt). clang-23 6-arg builtin:
// (g0 u32x4, g1 i32x8, g2 i32x4, g3 i32x4, i32x8, cpol).
// ---------------------------------------------------------------------------
__device__ __forceinline__ void tdm_load_tile(const float* gsrc, unsigned ldsByteOff) {
  unsigned long long ga = (unsigned long long)(uintptr_t)gsrc;
  u32x4 g0;
  g0[0] = 1u;                                  // count=1 (valid user descriptor)
  g0[1] = ldsByteOff;                          // lds_addr (bytes)
  g0[2] = (unsigned)ga;                        // global_addr[31:0]
  g0[3] = (unsigned)(ga >> 32) | 0x80000000u;  // global_addr[56:32] | type=2<<30
  i32x8 g1;
  g1[0] = 0x20000;                  // workgroup_mask=0, data_size=2 (4 bytes)
  g1[1] = (int)(128u << 16);        // tensor_dim0 = DIM (low 16 bits at [31:16])
  g1[2] = (int)(256u << 16);        // tensor_dim0 hi=0 | tensor_dim1 = NCENT
  g1[3] = (int)(128u << 16);        // tensor_dim1 hi=0 | tile_dim0 = DIM
  g1[4] = CTILE;                    // tile_dim1 = 16 rows, tile_dim2 = 0
  g1[5] = 128;                      // tensor_dim0_stride = DIM (low 32)
  g1[6] = 0;                        // stride hi | tensor_dim1_stride lo (2D: unused)
  g1[7] = 0;
  i32x4 z4 = {0, 0, 0, 0};          // groups 2/3 unused for 2D tiles
  i32x8 z8 = {0, 0, 0, 0, 0, 0, 0, 0};
  __builtin_amdgcn_tensor_load_to_lds(g0, g1, z4, z4, z8, 0);
}

// ---------------------------------------------------------------------------
// Kernel 1: reset workspace + precompute center norms. One block of 256.
// ---------------------------------------------------------------------------
__global__ void euc_init_kernel(const float* __restrict__ centers, ClusterWs* ws) {
  const int t = threadIdx.x;            // 0..255, one center per thread
  ws->argmin[t] = ~0ull;
  if (t == 0) ws->loss = 0.0f;
  const float* cp = centers + (size_t)t * DIM;
  float s = 0.0f;
#pragma unroll
  for (int k = 0; k < DIM; k += 4) {
    float4 v = *(const float4*)(cp + k);
    s += v.x * v.x + v.y * v.y + v.z * v.z + v.w * v.w;
  }
  ws->c2[t] = s;
}

// ---------------------------------------------------------------------------
// Kernel 2: fused distance GEMM (V_WMMA_F32_16X16X4_F32, B staged by TDM into
// double-buffered LDS) + row-min(loss) + column-argmin (packed u64 atomics).
// 256 threads = 8 waves, 16 rows/wave.
// ---------------------------------------------------------------------------
__global__ void __launch_bounds__(256)
euc_cluster_main(const float* __restrict__ embs, const float* __restrict__ centers,
                 int N, ClusterWs* __restrict__ ws) {
  __shared__ float sB[2][TILE_FLOATS];          // 2 x 8KB B-tile double buffer
  __shared__ unsigned long long sArg[NCENT];
  __shared__ float sLoss;

  const int t = threadIdx.x;
  sArg[t] = ~0ull;
  if (t == 0) sLoss = 0.0f;

  const int lane = t & 31;
  const int wave = t >> 5;
  const int half = lane >> 4;        // which K-pair this lane carries
  const int lrow = lane & 15;        // A-row (and B-column) owned by this lane
  const int koff = half * 2;
  const int rowBase = blockIdx.x * ROWS_PER_WG + wave * 16;

  // Kick off the DMA of the first B tile while we load A.
  if (wave == 0) tdm_load_tile(centers, (unsigned)(uintptr_t)&sB[0][0]);

  // ---- Load the full 16x128 A tile in WMMA striping; keep it in VGPRs. ----
  int arow = rowBase + lrow;
  if (arow >= N) arow = N - 1;       // tail clamp (duplicates can only tie in argmin)
  const float* ap = embs + (size_t)arow * DIM + koff;
  v2f a[32];
  float part = 0.0f;
#pragma unroll
  for (int kk = 0; kk < 32; ++kk) {
    a[kk] = *(const v2f*)(ap + kk * 4);
    part += a[kk].x * a[kk].x + a[kk].y * a[kk].y;
  }
  // ||e||^2 of row lrow: lanes l and l^16 hold complementary K-halves.
  const float e2row = part + __shfl_xor(part, 16, 32);

  // Accumulator register r of this lane maps to output row m = r + 8*half.
  float e2m[8];
  int growc[8];
  bool valid[8];
#pragma unroll
  for (int r = 0; r < 8; ++r) {
    e2m[r] = __shfl(e2row, r + half * 8, 32);
    int g = rowBase + r + half * 8;
    valid[r] = (g < N);
    growc[r] = valid[r] ? g : (N - 1);
  }

  float rowMin[8];
#pragma unroll
  for (int r = 0; r < 8; ++r) rowMin[r] = __builtin_inff();

  // ---- Sweep the 16 column tiles of 16 centers each. ----
  for (int tile = 0; tile < NTILES; ++tile) {
    if (wave == 0) __builtin_amdgcn_s_wait_tensorcnt(0);  // tile's DMA done
    __syncthreads();                                      // buffer visible to all
    // prefetch next tile into the other buffer (safe: everyone just left it)
    if (wave == 0 && tile + 1 < NTILES)
      tdm_load_tile(centers + (size_t)(tile + 1) * TILE_FLOATS,
                    (unsigned)(uintptr_t)&sB[(tile + 1) & 1][0]);

    // Pre-load the whole B tile into registers (ds_load_b64 clause), then run
    // the WMMA chain back-to-back on two interleaved accumulators.
    const float* bp = &sB[tile & 1][lrow * DIM + koff];
    v2f b[32];
#pragma unroll
    for (int kk = 0; kk < 32; ++kk) b[kk] = *(const v2f*)(bp + kk * 4);

    v8f acc0 = {0.f, 0.f, 0.f, 0.f, 0.f, 0.f, 0.f, 0.f};
    v8f acc1 = {0.f, 0.f, 0.f, 0.f, 0.f, 0.f, 0.f, 0.f};
#pragma unroll
    for (int kk = 0; kk < 32; kk += 2) {
      acc0 = __builtin_amdgcn_wmma_f32_16x16x4_f32(false, a[kk], false, b[kk],
                                                   (short)0, acc0, false, false);
      acc1 = __builtin_amdgcn_wmma_f32_16x16x4_f32(false, a[kk + 1], false, b[kk + 1],
                                                   (short)0, acc1, false, false);
    }
    v8f acc = acc0 + acc1;

    const int c = tile * CTILE;
    const float c2n = ws->c2[c + lrow];     // this lane's column n = c+lrow
    unsigned long long best = ~0ull;
#pragma unroll
    for (int r = 0; r < 8; ++r) {
      float d2 = fmaf(-2.0f, acc[r], e2m[r] + c2n);
      float dist = sqrtf(fmaxf(d2, 1e-12f));
      rowMin[r] = fminf(rowMin[r], dist);
      unsigned long long key =
          ((unsigned long long)__float_as_uint(dist) << 32) | (unsigned)growc[r];
      if (key < best) best = key;
    }
    // lanes l and l^16 cover the same column with complementary row halves
    unsigned int blo = (unsigned int)best, bhi = (unsigned int)(best >> 32);
    unsigned int olo = __shfl_xor(blo, 16, 32), ohi = __shfl_xor(bhi, 16, 32);
    unsigned long long other = ((unsigned long long)ohi << 32) | olo;
    if (other < best) best = other;
    if (lane < 16) atomicMin(&sArg[c + lrow], best);   // ds_min_u64 pre-reduction
  }

  // ---- Row-min across the 16 lanes of each half, sum valid rows for loss ----
  float ls = 0.0f;
#pragma unroll
  for (int r = 0; r < 8; ++r) {
    float m = rowMin[r];
    m = fminf(m, __shfl_xor(m, 1, 32));
    m = fminf(m, __shfl_xor(m, 2, 32));
    m = fminf(m, __shfl_xor(m, 4, 32));
    m = fminf(m, __shfl_xor(m, 8, 32));
    if (valid[r]) ls += m;
  }
  if (lrow == 0) atomicAdd(&sLoss, ls);   // lanes 0 and 16 of each wave
  __syncthreads();

  // one workgroup-level flush to global workspace
  atomicMin(&ws->argmin[t], sArg[t]);
  if (t == 0) atomicAdd(&ws->loss, sLoss);
}

// ---------------------------------------------------------------------------
// Kernel 3: extract rep ids, gather representative rows, emit loss.
// Output layout: [centers_out 256*128][rep_ids 256][loss 1]
// ---------------------------------------------------------------------------
__global__ void euc_finalize_kernel(const float* __restrict__ embs,
                                    const ClusterWs* __restrict__ ws,
                                    float* __restrict__ out) {
  const int k = threadIdx.x;  // 0..255
  const unsigned n = (unsigned)ws->argmin[k];
  const float4* src = (const float4*)(embs + (size_t)n * DIM);
  float4* dst = (float4*)(out + (size_t)k * DIM);
#pragma unroll
  for (int i = 0; i < DIM / 4; ++i) dst[i] = src[i];
  out[NCENT * DIM + k] = (float)n;
  if (k == 0) out[NCENT * DIM + NCENT] = ws->loss;
}

extern "C" void kernel_launch(void* const* d_in, const int* in_sizes, int n_in,
                              void* d_out, int out_size, void* d_ws, size_t ws_size,
                              hipStream_t stream) {
  const float* embs = (const float*)d_in[0];
  const float* centers = (const float*)d_in[1];
  float* out = (float*)d_out;
  ClusterWs* ws = (ClusterWs*)d_ws;
  const int N = in_sizes[0] / DIM;

  hipLaunchKernelGGL(euc_init_kernel, dim3(1), dim3(NCENT), 0, stream, centers, ws);
  const int grid = (N + ROWS_PER_WG - 1) / ROWS_PER_WG;
  hipLaunchKernelGGL(euc_cluster_main, dim3(grid), dim3(256), 0, stream,
                     embs, centers, N, ws);
  hipLaunchKernelGGL(euc_finalize_kernel, dim3(1), dim3(NCENT), 0, stream,
                     embs, ws, out);
}